// LiteNonLocalBlock_13907104104496
// MI455X (gfx1250) — compile-verified
//
#include <hip/hip_runtime.h>
#include <hip/hip_bf16.h>
#include <math.h>

typedef __attribute__((ext_vector_type(2))) float v2f;
typedef __attribute__((ext_vector_type(8))) float v8f;

// D(16x16,f32) = A(16x4,f32) x B(4x16,f32) + C   -- CDNA5 V_WMMA_F32_16X16X4_F32
__device__ __forceinline__ v8f wmma4(v2f a, v2f b, v8f c) {
    return __builtin_amdgcn_wmma_f32_16x16x4_f32(
        /*neg_a=*/false, a, /*neg_b=*/false, b,
        /*c_mod=*/(short)0, c, /*reuse_a=*/false, /*reuse_b=*/false);
}

// ---------------------------------------------------------------------------
// K1: 4x4 average pool.  x[8,256,128,128] -> xd[8,256,32,32] (= [b,c,1024])
// ---------------------------------------------------------------------------
__global__ void pool_kernel(const float* __restrict__ x, float* __restrict__ xd) {
    int idx = blockIdx.x * 256 + threadIdx.x;      // 8*256*1024 = 2^21 threads
    int n  = idx & 1023;
    int bc = idx >> 10;
    int cc = n & 31, r = n >> 5;
    const float* p = x + ((size_t)bc << 14) + (size_t)r * 4 * 128 + cc * 4;
    float s = 0.0f;
#pragma unroll
    for (int j = 0; j < 4; ++j) {
        float4 v = *(const float4*)(p + j * 128);
        s += v.x + v.y + v.z + v.w;
    }
    xd[idx] = s * (1.0f / 16.0f);
}

// ---------------------------------------------------------------------------
// K2: theta/phi/g = W[16,256] @ xd[256,1024] + bias.  One wave per
// (b, 16-col tile, projection); 64 k-steps of V_WMMA_F32_16X16X4_F32.
// ---------------------------------------------------------------------------
__global__ void proj_kernel(const float* __restrict__ xd,
                            const float* __restrict__ tw, const float* __restrict__ tb,
                            const float* __restrict__ pw, const float* __restrict__ pb,
                            const float* __restrict__ gw, const float* __restrict__ gb,
                            float* __restrict__ theta, float* __restrict__ phi,
                            float* __restrict__ gout) {
    int unit = blockIdx.x * (blockDim.x >> 5) + (threadIdx.x >> 5);  // 1536 units
    int lane = threadIdx.x & 31;
    int proj = unit % 3;
    int nt   = (unit / 3) & 63;
    int b    = unit / 192;
    const float* W; const float* bias; float* dst;
    if (proj == 0)      { W = tw; bias = tb; dst = theta; }
    else if (proj == 1) { W = pw; bias = pb; dst = phi;   }
    else                { W = gw; bias = gb; dst = gout;  }

    int lo = lane & 15, kh = lane >> 4;
    const float* xb = xd + (size_t)b * 262144 + nt * 16;
    v8f acc = {};
    for (int ks = 0; ks < 64; ++ks) {
        int kb = ks * 4 + 2 * kh;
        v2f a;  a.x  = W[lo * 256 + kb];            a.y  = W[lo * 256 + kb + 1];
        v2f bb; bb.x = xb[(size_t)kb * 1024 + lo];  bb.y = xb[(size_t)(kb + 1) * 1024 + lo];
        acc = wmma4(a, bb, acc);
    }
    float* d = dst + (size_t)b * 16384 + nt * 16;
#pragma unroll
    for (int r = 0; r < 8; ++r) {
        int m = r + 8 * kh;
        d[(size_t)m * 1024 + lo] = acc[r] + bias[m];
    }
}

// ---------------------------------------------------------------------------
// K3: S[b,n,m] = sum_c theta[b,c,n]*phi[b,c,m].  One wave per 16x16 tile,
// K=16 -> 4 WMMAs.
// ---------------------------------------------------------------------------
__global__ void scores_kernel(const float* __restrict__ theta,
                              const float* __restrict__ phi,
                              float* __restrict__ att) {
    int unit = blockIdx.x * (blockDim.x >> 5) + (threadIdx.x >> 5);  // 8*64*64
    int lane = threadIdx.x & 31;
    int mt = unit & 63;
    int nt = (unit >> 6) & 63;
    int b  = unit >> 12;
    const float* T = theta + (size_t)b * 16384;
    const float* P = phi   + (size_t)b * 16384;
    int lo = lane & 15, kh = lane >> 4;
    v8f acc = {};
#pragma unroll
    for (int ks = 0; ks < 4; ++ks) {
        int kb = ks * 4 + 2 * kh;
        v2f a;  a.x  = T[(size_t)kb * 1024 + nt * 16 + lo];
                a.y  = T[(size_t)(kb + 1) * 1024 + nt * 16 + lo];
        v2f bb; bb.x = P[(size_t)kb * 1024 + mt * 16 + lo];
                bb.y = P[(size_t)(kb + 1) * 1024 + mt * 16 + lo];
        acc = wmma4(a, bb, acc);
    }
    float* S = att + (size_t)b * 1048576;
#pragma unroll
    for (int r = 0; r < 8; ++r) {
        int n = nt * 16 + r + 8 * kh;
        int m = mt * 16 + lo;
        S[(size_t)n * 1024 + m] = acc[r];
    }
}

// ---------------------------------------------------------------------------
// K4: row softmax over att[row, 0:1024].  One 256-thread block per row.
// ---------------------------------------------------------------------------
__global__ void softmax_kernel(float* __restrict__ att) {
    __shared__ float sd[256];
    float* p = att + (size_t)blockIdx.x * 1024;
    int t = threadIdx.x;
    float4 v = ((float4*)p)[t];
    float m = fmaxf(fmaxf(v.x, v.y), fmaxf(v.z, v.w));
    sd[t] = m; __syncthreads();
    for (int s = 128; s > 0; s >>= 1) { if (t < s) sd[t] = fmaxf(sd[t], sd[t + s]); __syncthreads(); }
    m = sd[0]; __syncthreads();
    v.x = __expf(v.x - m); v.y = __expf(v.y - m);
    v.z = __expf(v.z - m); v.w = __expf(v.w - m);
    sd[t] = v.x + v.y + v.z + v.w; __syncthreads();
    for (int s = 128; s > 0; s >>= 1) { if (t < s) sd[t] += sd[t + s]; __syncthreads(); }
    float inv = 1.0f / sd[0];
    v.x *= inv; v.y *= inv; v.z *= inv; v.w *= inv;
    ((float4*)p)[t] = v;
}

// ---------------------------------------------------------------------------
// K5: out[b,c,n] = sum_m g[b,c,m]*att[b,n,m].  K=1024 split across 4 waves
// per block (one block per (b, n-tile)), LDS reduction.
// ---------------------------------------------------------------------------
__global__ void attn_out_kernel(const float* __restrict__ g,
                                const float* __restrict__ att,
                                float* __restrict__ outbuf) {
    __shared__ float red[4][8][32];
    int nt = blockIdx.x & 63;
    int b  = blockIdx.x >> 6;
    int wave = threadIdx.x >> 5, lane = threadIdx.x & 31;
    int lo = lane & 15, kh = lane >> 4;
    const float* G = g   + (size_t)b * 16384;
    const float* A = att + (size_t)b * 1048576;
    v8f acc = {};
    for (int ks = 0; ks < 64; ++ks) {
        int kb = wave * 256 + ks * 4 + 2 * kh;
        v2f a;  a.x  = G[(size_t)lo * 1024 + kb];
                a.y  = G[(size_t)lo * 1024 + kb + 1];
        v2f bb; bb.x = A[(size_t)(nt * 16 + lo) * 1024 + kb];
                bb.y = A[(size_t)(nt * 16 + lo) * 1024 + kb + 1];
        acc = wmma4(a, bb, acc);
    }
#pragma unroll
    for (int r = 0; r < 8; ++r) red[wave][r][lane] = acc[r];
    __syncthreads();
    if (wave == 0) {
        float* D = outbuf + (size_t)b * 16384 + nt * 16;
#pragma unroll
        for (int r = 0; r < 8; ++r) {
            float s = red[0][r][lane] + red[1][r][lane] + red[2][r][lane] + red[3][r][lane];
            int c = r + 8 * kh;
            D[(size_t)c * 1024 + lo] = s;
        }
    }
}

// ---------------------------------------------------------------------------
// K6: y[b,co,n] = out_w[co,0:16] @ out[b,0:16,n] + out_b[co]
// ---------------------------------------------------------------------------
__global__ void outconv_kernel(const float* __restrict__ outbuf,
                               const float* __restrict__ ow,
                               const float* __restrict__ ob,
                               float* __restrict__ y) {
    int unit = blockIdx.x * (blockDim.x >> 5) + (threadIdx.x >> 5);  // 8*16*64
    int lane = threadIdx.x & 31;
    int nt = unit & 63;
    int ct = (unit >> 6) & 15;
    int b  = unit >> 10;
    int lo = lane & 15, kh = lane >> 4;
    const float* O = outbuf + (size_t)b * 16384 + nt * 16;
    v8f acc = {};
#pragma unroll
    for (int ks = 0; ks < 4; ++ks) {
        int kb = ks * 4 + 2 * kh;
        v2f a;  a.x  = ow[(ct * 16 + lo) * 16 + kb];
                a.y  = ow[(ct * 16 + lo) * 16 + kb + 1];
        v2f bb; bb.x = O[(size_t)kb * 1024 + lo];
                bb.y = O[(size_t)(kb + 1) * 1024 + lo];
        acc = wmma4(a, bb, acc);
    }
    float* Y = y + (size_t)b * 262144 + (size_t)ct * 16 * 1024 + nt * 16;
#pragma unroll
    for (int r = 0; r < 8; ++r) {
        int co = r + 8 * kh;
        Y[(size_t)co * 1024 + lo] = acc[r] + ob[ct * 16 + co];
    }
}

// ---------------------------------------------------------------------------
// K7: GroupNorm stats per (b, group): 32 channels x 1024 px contiguous.
// ---------------------------------------------------------------------------
__global__ void gnstats_kernel(const float* __restrict__ y, float* __restrict__ stats) {
    __shared__ float ss[256], sq[256];
    const float* p = y + (size_t)blockIdx.x * 32768;   // 64 blocks: b*8+g
    int t = threadIdx.x;
    float s = 0.0f, q = 0.0f;
#pragma unroll
    for (int i = 0; i < 32; ++i) {
        float4 v = ((const float4*)p)[t + i * 256];
        s += v.x + v.y + v.z + v.w;
        q += v.x * v.x + v.y * v.y + v.z * v.z + v.w * v.w;
    }
    ss[t] = s; sq[t] = q; __syncthreads();
    for (int st = 128; st > 0; st >>= 1) {
        if (t < st) { ss[t] += ss[t + st]; sq[t] += sq[t + st]; }
        __syncthreads();
    }
    if (t == 0) {
        float mean = ss[0] * (1.0f / 32768.0f);
        float var  = sq[0] * (1.0f / 32768.0f) - mean * mean;
        stats[blockIdx.x * 2]     = mean;
        stats[blockIdx.x * 2 + 1] = rsqrtf(var + 1e-5f);
    }
}

// ---------------------------------------------------------------------------
// K8: fused GN-normalize + affine + bilinear 4x upsample + residual,
// vectorized: one thread -> 4 consecutive output pixels (one src interval).
// For xx = 4k+j the src column weights are fixed:
//   j=0: .375*L+.625*M   j=1: .125*L+.875*M
//   j=2: .875*M+.125*R   j=3: .625*M+.375*R   (L=k-1, M=k, R=k+1, clamped)
// GN affine is linear per channel -> fold (mean,rstd,gn_w,gn_b,0.5) into
// a single per-channel alpha/beta applied after interpolation.
// ---------------------------------------------------------------------------
__global__ void up_kernel(const float* __restrict__ x, const float* __restrict__ y,
                          const float* __restrict__ stats,
                          const float* __restrict__ gw, const float* __restrict__ gb,
                          float* __restrict__ out) {
    int idx = blockIdx.x * 256 + threadIdx.x;          // 2^23 threads
    int k  = idx & 31;                                  // src column group
    int yy = (idx >> 5) & 127;
    int c  = (idx >> 12) & 255;
    int b  = idx >> 20;

    // vertical weights
    float sy  = (yy + 0.5f) * 0.25f - 0.5f;
    float fy0 = floorf(sy);
    float wy  = sy - fy0;
    int iy0 = max(0, min(31, (int)fy0));
    int iy1 = max(0, min(31, (int)fy0 + 1));

    int cL = max(0, k - 1);
    int cR = min(31, k + 1);

    const float* P = y + ((size_t)(b * 256 + c)) * 1024;
    const float* r0 = P + iy0 * 32;
    const float* r1 = P + iy1 * 32;
    float L0 = r0[cL], M0 = r0[k], R0 = r0[cR];
    float L1 = r1[cL], M1 = r1[k], R1 = r1[cR];

    // vertical blend of the 3 source columns, then horizontal fixed weights
    float Lv = L0 + (L1 - L0) * wy;
    float Mv = M0 + (M1 - M0) * wy;
    float Rv = R0 + (R1 - R0) * wy;

    float4 bil;
    bil.x = 0.375f * Lv + 0.625f * Mv;
    bil.y = 0.125f * Lv + 0.875f * Mv;
    bil.z = 0.875f * Mv + 0.125f * Rv;
    bil.w = 0.625f * Mv + 0.375f * Rv;

    int grp = b * 8 + (c >> 5);
    float mean = stats[grp * 2], rstd = stats[grp * 2 + 1];
    float alpha = 0.5f * rstd * gw[c];
    float beta  = 0.5f * (gb[c] - mean * rstd * gw[c]);

    float4 xv = ((const float4*)x)[idx];
    float4 o;
    o.x = fmaf(bil.x, alpha, xv.x + beta);
    o.y = fmaf(bil.y, alpha, xv.y + beta);
    o.z = fmaf(bil.z, alpha, xv.z + beta);
    o.w = fmaf(bil.w, alpha, xv.w + beta);
    ((float4*)out)[idx] = o;
}

// ---------------------------------------------------------------------------
extern "C" void kernel_launch(void* const* d_in, const int* in_sizes, int n_in,
                              void* d_out, int out_size, void* d_ws, size_t ws_size,
                              hipStream_t stream) {
    const float* x   = (const float*)d_in[0];
    const float* tw  = (const float*)d_in[1];
    const float* tb  = (const float*)d_in[2];
    const float* pw  = (const float*)d_in[3];
    const float* pb  = (const float*)d_in[4];
    const float* gw_ = (const float*)d_in[5];
    const float* gb_ = (const float*)d_in[6];
    const float* ow  = (const float*)d_in[7];
    const float* ob  = (const float*)d_in[8];
    const float* gnw = (const float*)d_in[9];
    const float* gnb = (const float*)d_in[10];
    float* out = (float*)d_out;

    float* ws     = (float*)d_ws;
    float* xd     = ws;                    // 8*256*1024        = 2,097,152 f
    float* theta  = xd     + 2097152;      // 8*16*1024         =   131,072 f
    float* phi    = theta  + 131072;
    float* gbuf   = phi    + 131072;
    float* att    = gbuf   + 131072;       // 8*1024*1024       = 8,388,608 f
    float* outbuf = att    + 8388608;      // 8*16*1024
    float* y      = outbuf + 131072;       // 8*256*1024
    float* stats  = y      + 2097152;      // 64*2

    pool_kernel    <<<8192,  256, 0, stream>>>(x, xd);
    proj_kernel    <<<192,   256, 0, stream>>>(xd, tw, tb, pw, pb, gw_, gb_,
                                               theta, phi, gbuf);
    scores_kernel  <<<4096,  256, 0, stream>>>(theta, phi, att);
    softmax_kernel <<<8192,  256, 0, stream>>>(att);
    attn_out_kernel<<<512,   128, 0, stream>>>(gbuf, att, outbuf);
    outconv_kernel <<<1024,  256, 0, stream>>>(outbuf, ow, ob, y);
    gnstats_kernel <<<64,    256, 0, stream>>>(y, stats);
    up_kernel      <<<32768, 256, 0, stream>>>(x, y, stats, gnw, gnb, out);
}